// TCRcube_35828617183361
// MI455X (gfx1250) — compile-verified
//
#include <hip/hip_runtime.h>

// ---------------------------------------------------------------------------
// Problem constants (from reference)
// ---------------------------------------------------------------------------
#define BB   32
#define LA_  24
#define LB_  24
#define LP_  14
#define LM_  180
#define EE   1280
#define DD   32

typedef __attribute__((ext_vector_type(2))) float v2f;
typedef __attribute__((ext_vector_type(8))) float v8f;

// ---------------------------------------------------------------------------
// Stage 0: zero the 32-float output (harness poisons d_out)
// ---------------------------------------------------------------------------
__global__ void zero_out_kernel(float* __restrict__ out, int n) {
    int i = blockIdx.x * blockDim.x + threadIdx.x;
    if (i < n) out[i] = 0.0f;
}

// ---------------------------------------------------------------------------
// Stage 1: projection GEMMs  Y[M,32] = X[M,1280] @ W[1280,32] + bias
//
// One wave per 64-row super-tile (4 M-tiles of 16), 8 f32 accumulators
// (4 M-tiles x 2 N-tiles), V_WMMA_F32_16X16X4_F32 with K-step 4.
// B operand (W columns) loaded once per K-step and shared by all 4 M-tiles:
// 8 VMEM loads -> 8 WMMAs per K-step.
//
// A (16x4 f32) ISA layout: lanes 0-15 hold M=0..15; VGPR0/1 = K pair,
// lane-half selects K offset {0,1} vs {2,3}.  B (4x16) mirrored: lanes = N,
// VGPR pair = consecutive K, lane-half selects K offset.
// C/D: lane = N (both halves), VGPR r -> M = r + 8*(lane>>4).
// ---------------------------------------------------------------------------
__global__ __launch_bounds__(32) void proj_wmma_kernel(
    const float* __restrict__ Xa, const float* __restrict__ Xb,
    const float* __restrict__ Xp, const float* __restrict__ Xm,
    const float* __restrict__ Wa, const float* __restrict__ Wb,
    const float* __restrict__ Wp, const float* __restrict__ Wm,
    const float* __restrict__ ba, const float* __restrict__ bbias,
    const float* __restrict__ bp, const float* __restrict__ bm,
    float* __restrict__ Ya, float* __restrict__ Yb,
    float* __restrict__ Yp, float* __restrict__ Ym)
{
    // super-tile table (64 rows each): a:12, b:12, p:7, m:90  -> 121 blocks
    int t = blockIdx.x;
    const float *X, *W, *bias;
    float* Y;
    int stile;
    if (t < 12)      { X = Xa; W = Wa; bias = ba;    Y = Ya; stile = t;      }
    else if (t < 24) { X = Xb; W = Wb; bias = bbias; Y = Yb; stile = t - 12; }
    else if (t < 31) { X = Xp; W = Wp; bias = bp;    Y = Yp; stile = t - 24; }
    else             { X = Xm; W = Wm; bias = bm;    Y = Ym; stile = t - 31; }

    const int lane  = threadIdx.x;      // 0..31
    const int lrow  = lane & 15;        // M within tile for A; N for B/C
    const int khalf = lane >> 4;        // selects K offset pair

    const int row0 = stile * 64 + lrow;
    const float* xr0 = X + (size_t)(row0)      * EE;
    const float* xr1 = X + (size_t)(row0 + 16) * EE;
    const float* xr2 = X + (size_t)(row0 + 32) * EE;
    const float* xr3 = X + (size_t)(row0 + 48) * EE;

    v8f acc0[4] = {};   // N = 0..15, M-tiles 0..3
    v8f acc1[4] = {};   // N = 16..31, M-tiles 0..3

    for (int k = 0; k < EE; k += 4) {
        const int ka = k + 2 * khalf;

        v2f a0, a1, a2, a3;
        a0.x = xr0[ka]; a0.y = xr0[ka + 1];
        a1.x = xr1[ka]; a1.y = xr1[ka + 1];
        a2.x = xr2[ka]; a2.y = xr2[ka + 1];
        a3.x = xr3[ka]; a3.y = xr3[ka + 1];

        v2f b0, b1;
        b0.x = W[(size_t)ka * DD + lrow];
        b0.y = W[(size_t)(ka + 1) * DD + lrow];
        b1.x = W[(size_t)ka * DD + lrow + 16];
        b1.y = W[(size_t)(ka + 1) * DD + lrow + 16];

        acc0[0] = __builtin_amdgcn_wmma_f32_16x16x4_f32(false, a0, false, b0, (short)0, acc0[0], false, false);
        acc1[0] = __builtin_amdgcn_wmma_f32_16x16x4_f32(false, a0, false, b1, (short)0, acc1[0], false, false);
        acc0[1] = __builtin_amdgcn_wmma_f32_16x16x4_f32(false, a1, false, b0, (short)0, acc0[1], false, false);
        acc1[1] = __builtin_amdgcn_wmma_f32_16x16x4_f32(false, a1, false, b1, (short)0, acc1[1], false, false);
        acc0[2] = __builtin_amdgcn_wmma_f32_16x16x4_f32(false, a2, false, b0, (short)0, acc0[2], false, false);
        acc1[2] = __builtin_amdgcn_wmma_f32_16x16x4_f32(false, a2, false, b1, (short)0, acc1[2], false, false);
        acc0[3] = __builtin_amdgcn_wmma_f32_16x16x4_f32(false, a3, false, b0, (short)0, acc0[3], false, false);
        acc1[3] = __builtin_amdgcn_wmma_f32_16x16x4_f32(false, a3, false, b1, (short)0, acc1[3], false, false);
    }

    // Store with bias: lane l -> N = l&15 (+16 for acc1); VGPR r -> M = r + 8*(l>>4)
    const int n0    = lrow;
    const float bv0 = bias[n0];
    const float bv1 = bias[n0 + 16];
#pragma unroll
    for (int mt = 0; mt < 4; ++mt) {
        const int mbase = stile * 64 + mt * 16 + 8 * khalf;
#pragma unroll
        for (int r = 0; r < 8; ++r) {
            const int m = mbase + r;
            Y[(size_t)m * DD + n0]      = acc0[mt][r] + bv0;
            Y[(size_t)m * DD + n0 + 16] = acc1[mt][r] + bv1;
        }
    }
}

// ---------------------------------------------------------------------------
// Stage 2: cube + LayerNorm + ReLU + dot(Wo) + masked average.
// One block per (b, i, branch): 192 threads (6 waves), thread k owns m-row k.
// ap[j][e] = x_i[e]*p_j[e] staged in LDS, reused by all 180 k-threads.
// Exact two-pass variance to match reference mean((v-mu)^2).
// ---------------------------------------------------------------------------
__global__ __launch_bounds__(192) void cube_kernel(
    const float* __restrict__ aproj, const float* __restrict__ bproj,
    const float* __restrict__ pproj, const float* __restrict__ mproj,
    const float* __restrict__ amask, const float* __restrict__ bmask,
    const float* __restrict__ pmask, const float* __restrict__ mmask,
    const float* __restrict__ lens,
    const float* __restrict__ ga,  const float* __restrict__ bea,
    const float* __restrict__ Wao, const float* __restrict__ bao,
    const float* __restrict__ gb,  const float* __restrict__ beb,
    const float* __restrict__ Wbo, const float* __restrict__ bbo,
    float* __restrict__ out)
{
    const int blk    = blockIdx.x;            // B * (LA+LB) blocks
    const int b      = blk / (LA_ + LB_);
    const int r      = blk % (LA_ + LB_);
    const int branch = (r >= LA_) ? 1 : 0;
    const int i      = branch ? (r - LA_) : r;

    const float* xrow = branch ? (bproj + (size_t)(b * LB_ + i) * DD)
                               : (aproj + (size_t)(b * LA_ + i) * DD);
    const float* g    = branch ? gb  : ga;
    const float* be   = branch ? beb : bea;
    const float* Wo   = branch ? Wbo : Wao;
    const float  bo   = branch ? bbo[0] : bao[0];
    const float  im   = branch ? bmask[b * LB_ + i] : amask[b * LA_ + i];
    const float  denom = lens[b * 4 + (branch ? 1 : 0)] * lens[b * 4 + 2] * lens[b * 4 + 3];

    __shared__ float ap_s[LP_][DD];
    __shared__ float pm_s[LP_];
    __shared__ float red[256];

    const int tid = threadIdx.x;

    for (int idx = tid; idx < LP_ * DD; idx += blockDim.x) {
        const int j = idx / DD, e = idx % DD;
        ap_s[j][e] = xrow[e] * pproj[(size_t)(b * LP_ + j) * DD + e];
    }
    if (tid < LP_) pm_s[tid] = pmask[b * LP_ + tid];
    __syncthreads();

    // uniform per-block constants -> compiler scalarizes to SGPRs
    float G[DD], Be[DD], Wv[DD];
#pragma unroll
    for (int e = 0; e < DD; ++e) { G[e] = g[e]; Be[e] = be[e]; Wv[e] = Wo[e]; }

    float acc = 0.0f;
    const int k = tid;
    if (k < LM_) {
        float Mv[DD];
        const float* mrow = mproj + (size_t)(b * LM_ + k) * DD;
#pragma unroll
        for (int e = 0; e < DD; ++e) Mv[e] = mrow[e];
        const float km = mmask[b * LM_ + k];

        float acck = 0.0f;
        for (int j = 0; j < LP_; ++j) {
            float v[DD];
            float sum = 0.0f;
#pragma unroll
            for (int e = 0; e < DD; ++e) {
                const float t = ap_s[j][e] * Mv[e];
                v[e] = t;
                sum += t;
            }
            const float mu = sum * (1.0f / 32.0f);
            float sq = 0.0f;
#pragma unroll
            for (int e = 0; e < DD; ++e) {
                const float d = v[e] - mu;
                sq = fmaf(d, d, sq);
            }
            const float rs = rsqrtf(sq * (1.0f / 32.0f) + 1e-5f);
            float dot = 0.0f;
#pragma unroll
            for (int e = 0; e < DD; ++e) {
                const float t = (v[e] - mu) * rs;
                float h = fmaf(t, G[e], Be[e]);
                h = fmaxf(h, 0.0f);
                dot = fmaf(h, Wv[e], dot);
            }
            acck += (dot + bo) * pm_s[j];
        }
        acc = acck * km;
    }

    // block reduction (pad 192 -> 256)
    red[tid] = acc;
    if (tid < 64) red[192 + tid] = 0.0f;
    __syncthreads();
    for (int s = 128; s > 0; s >>= 1) {
        if (tid < s) red[tid] += red[tid + s];
        __syncthreads();
    }
    if (tid == 0) atomicAdd(&out[b], red[0] * im / denom);
}

// ---------------------------------------------------------------------------
// Launch
// ---------------------------------------------------------------------------
extern "C" void kernel_launch(void* const* d_in, const int* in_sizes, int n_in,
                              void* d_out, int out_size, void* d_ws, size_t ws_size,
                              hipStream_t stream) {
    (void)in_sizes; (void)n_in; (void)ws_size;

    const float* CDR3a   = (const float*)d_in[0];
    const float* CDR3b   = (const float*)d_in[1];
    const float* peptide = (const float*)d_in[2];
    const float* mhc     = (const float*)d_in[3];
    const float* amask   = (const float*)d_in[4];
    const float* bmask   = (const float*)d_in[5];
    const float* pmask   = (const float*)d_in[6];
    const float* mmask   = (const float*)d_in[7];
    const float* lens    = (const float*)d_in[8];
    const float* Wa  = (const float*)d_in[9];
    const float* ba  = (const float*)d_in[10];
    const float* Wb  = (const float*)d_in[11];
    const float* bb  = (const float*)d_in[12];
    const float* Wp  = (const float*)d_in[13];
    const float* bp  = (const float*)d_in[14];
    const float* Wm  = (const float*)d_in[15];
    const float* bm  = (const float*)d_in[16];
    const float* ga  = (const float*)d_in[17];
    const float* bea = (const float*)d_in[18];
    const float* Wao = (const float*)d_in[19];
    const float* bao = (const float*)d_in[20];
    const float* gb  = (const float*)d_in[21];
    const float* beb = (const float*)d_in[22];
    const float* Wbo = (const float*)d_in[23];
    const float* bbo = (const float*)d_in[24];

    float* out = (float*)d_out;
    float* ws  = (float*)d_ws;

    // workspace layout (f32 elements)
    float* aproj = ws;                       // 768  * 32 = 24576
    float* bproj = ws + 24576;               // 768  * 32
    float* pproj = ws + 49152;               // 448  * 32 = 14336
    float* mproj = ws + 63488;               // 5760 * 32 = 184320

    zero_out_kernel<<<1, 32, 0, stream>>>(out, out_size);

    // 12 + 12 + 7 + 90 = 121 super-tiles of 64 rows, one wave each
    proj_wmma_kernel<<<121, 32, 0, stream>>>(
        CDR3a, CDR3b, peptide, mhc,
        Wa, Wb, Wp, Wm,
        ba, bb, bp, bm,
        aproj, bproj, pproj, mproj);

    cube_kernel<<<BB * (LA_ + LB_), 192, 0, stream>>>(
        aproj, bproj, pproj, mproj,
        amask, bmask, pmask, mmask, lens,
        ga, bea, Wao, bao,
        gb, beb, Wbo, bbo,
        out);
}